// G_Align_32744830665189
// MI455X (gfx1250) — compile-verified
//
#include <hip/hip_runtime.h>
#include <hip/hip_bf16.h>
#include <stdint.h>

typedef __attribute__((ext_vector_type(2))) float v2f;
typedef __attribute__((ext_vector_type(8))) float v8f;

#define FEATS   256
#define SCAN_T  1024

// ---------------------------------------------------------------------------
// h[i][j] = sum_k x[i][k] * W[j][k]   (x @ W^T), fp32 WMMA 16x16x4 path.
// One wave computes one 16x16 tile of h; block = 8 waves = 8 N-tiles.
// (Verified: pipelined v_wmma_f32_16x16x4_f32 + global_load_b64 codegen.)
// ---------------------------------------------------------------------------
__global__ __launch_bounds__(256) void gcn_gemm_xWT(const float* __restrict__ x,
                                                    const float* __restrict__ W,
                                                    float* __restrict__ h) {
    const int waveInBlk = threadIdx.x >> 5;          // 0..7
    const int lane      = threadIdx.x & 31;
    const int laneLo    = lane & 15;
    const int koff      = (lane >> 4) << 1;          // 0 or 2

    const int mbase = blockIdx.x << 4;                       // M tile
    const int nbase = ((blockIdx.y << 3) + waveInBlk) << 4;  // N tile

    const float* xrow = x + (size_t)(mbase + laneLo) * FEATS + koff;
    const float* wrow = W + (size_t)(nbase + laneLo) * FEATS + koff;

    v8f acc = {};
#pragma unroll 8
    for (int k = 0; k < FEATS; k += 4) {
        v2f a = *(const v2f*)(xrow + k);
        v2f b = *(const v2f*)(wrow + k);
        acc = __builtin_amdgcn_wmma_f32_16x16x4_f32(
            /*neg_a=*/false, a, /*neg_b=*/false, b,
            /*c_mod=*/(short)0, acc, /*reuse_a=*/false, /*reuse_b=*/false);
    }

    const int rowAdd = (lane >> 4) << 3;             // 0 or 8
    float* out = h + (size_t)(mbase + rowAdd) * FEATS + nbase + laneLo;
#pragma unroll
    for (int r = 0; r < 8; ++r) out[(size_t)r * FEATS] = acc[r];
}

// ------------------------- CSR construction -------------------------------

__global__ __launch_bounds__(256) void gcn_zero_i32(int* __restrict__ p, int n) {
    int i = blockIdx.x * 256 + threadIdx.x;
    if (i < n) p[i] = 0;
}

// counts[row]++ for every edge (integer atomics only: 3.2M, not 819M f32)
__global__ __launch_bounds__(256) void gcn_count(const long long* __restrict__ edges,
                                                 int* __restrict__ counts, int nEdges) {
    int e = blockIdx.x * 256 + threadIdx.x;
    if (e < nEdges) atomicAdd(&counts[(int)edges[(size_t)2 * e]], 1);
}

// Exclusive prefix sum over counts -> offsets (and cursor copy).
// Single 1024-thread workgroup, LDS Hillis-Steele scan, serial carry.
__global__ __launch_bounds__(SCAN_T) void gcn_scan(const int* __restrict__ counts,
                                                   int* __restrict__ offsets,
                                                   int* __restrict__ cursor, int n) {
    __shared__ int buf[SCAN_T];
    __shared__ int carry;
    const int t = threadIdx.x;
    if (t == 0) carry = 0;
    __syncthreads();
    for (int base = 0; base < n; base += SCAN_T) {
        const int idx = base + t;
        const int v = (idx < n) ? counts[idx] : 0;
        buf[t] = v;
        __syncthreads();
        for (int o = 1; o < SCAN_T; o <<= 1) {
            int xprev = (t >= o) ? buf[t - o] : 0;
            __syncthreads();
            buf[t] += xprev;
            __syncthreads();
        }
        const int incl = buf[t];
        const int c0 = carry;                     // stable: last write was pre-barrier
        if (idx < n) {
            int excl = c0 + incl - v;
            offsets[idx] = excl;
            cursor[idx]  = excl;
        }
        __syncthreads();
        if (t == SCAN_T - 1) carry = c0 + incl;
        __syncthreads();
    }
}

// Scatter each edge's (col, w) into its row's CSR segment.
__global__ __launch_bounds__(256) void gcn_fill(const long long* __restrict__ edges,
                                                const float* __restrict__ ew,
                                                int* __restrict__ cursor,
                                                int* __restrict__ colS,
                                                float* __restrict__ wS, int nEdges) {
    int e = blockIdx.x * 256 + threadIdx.x;
    if (e >= nEdges) return;
    int r = (int)edges[(size_t)2 * e];
    int c = (int)edges[(size_t)2 * e + 1];
    int slot = atomicAdd(&cursor[r], 1);
    colS[slot] = c;
    wS[slot]   = ew[e];
}

// ---------------------------------------------------------------------------
// Gather aggregation + fused tanh: one block per node, thread t owns feature
// t. (col, w) list staged through LDS; h rows gathered from L2-resident h.
// out[node][t] = tanh( sum_e w_e * h[col_e][t] )   — zero atomics.
// ---------------------------------------------------------------------------
__global__ __launch_bounds__(256) void gcn_aggregate(const int* __restrict__ offsets,
                                                     const int* __restrict__ counts,
                                                     const int* __restrict__ colS,
                                                     const float* __restrict__ wS,
                                                     const float* __restrict__ h,
                                                     float* __restrict__ out) {
    __shared__ int   s_col[256];
    __shared__ float s_w[256];
    const int node = blockIdx.x;
    const int t    = threadIdx.x;
    const int start = offsets[node];
    const int cnt   = counts[node];

    float acc0 = 0.0f, acc1 = 0.0f;
    for (int base = 0; base < cnt; base += 256) {
        const int m = min(cnt - base, 256);
        if (t < m) {
            s_col[t] = colS[start + base + t];
            s_w[t]   = wS[start + base + t];
        }
        __syncthreads();
        int i = 0;
        for (; i + 2 <= m; i += 2) {
            acc0 = fmaf(s_w[i + 0], h[(size_t)s_col[i + 0] * FEATS + t], acc0);
            acc1 = fmaf(s_w[i + 1], h[(size_t)s_col[i + 1] * FEATS + t], acc1);
        }
        if (i < m)
            acc0 = fmaf(s_w[i], h[(size_t)s_col[i] * FEATS + t], acc0);
        __syncthreads();
    }
    out[(size_t)node * FEATS + t] = tanhf(acc0 + acc1);
}

__global__ __launch_bounds__(256) void gcn_copy(const float* __restrict__ s,
                                                float* __restrict__ d, size_t n) {
    size_t i = (size_t)blockIdx.x * 256 + threadIdx.x;
    if (i < n) d[i] = s[i];
}

extern "C" void kernel_launch(void* const* d_in, const int* in_sizes, int n_in,
                              void* d_out, int out_size, void* d_ws, size_t ws_size,
                              hipStream_t stream) {
    const float*     x     = (const float*)d_in[0];      // (N, 256)  f32
    const long long* edges = (const long long*)d_in[1];  // (E, 2)    i64
    const float*     ew    = (const float*)d_in[2];      // (E,)      f32
    const float*     W0    = (const float*)d_in[3];      // (256,256) f32
    const float*     W1    = (const float*)d_in[4];      // (256,256) f32
    (void)n_in; (void)out_size; (void)ws_size;

    const int    N      = in_sizes[0] / FEATS;           // 100000
    const int    nEdges = in_sizes[1] / 2;               // 3,200,000
    const size_t slab   = (size_t)N * FEATS;             // 25.6M floats

    // workspace layout
    float* h       = (float*)d_ws;                       // slab floats (102.4 MB)
    float* wS      = h + slab;                           // nEdges floats
    int*   colS    = (int*)(wS + nEdges);                // nEdges ints
    int*   counts  = colS + nEdges;                      // N ints
    int*   offsets = counts + N;                         // N ints
    int*   cursor  = offsets + N;                        // N ints

    float* out0 = (float*)d_out;
    float* out1 = out0 + slab;
    float* out2 = out0 + 2 * slab;

    const dim3 gemmGrid(N / 16, 2);
    const int  elemBlocks = (int)((slab + 255) / 256);
    const int  edgeBlocks = (nEdges + 255) / 256;
    const int  nodeBlocks = (N + 255) / 256;

    // out0 = x
    gcn_copy<<<elemBlocks, 256, 0, stream>>>(x, out0, slab);

    // ---- CSR build (once per call, reused by both layers) ----
    gcn_zero_i32<<<nodeBlocks, 256, 0, stream>>>(counts, N);
    gcn_count<<<edgeBlocks, 256, 0, stream>>>(edges, counts, nEdges);
    gcn_scan<<<1, SCAN_T, 0, stream>>>(counts, offsets, cursor, N);
    gcn_fill<<<edgeBlocks, 256, 0, stream>>>(edges, ew, cursor, colS, wS, nEdges);

    // ---- layer 1 ----
    gcn_gemm_xWT<<<gemmGrid, 256, 0, stream>>>(x, W0, h);
    gcn_aggregate<<<N, 256, 0, stream>>>(offsets, counts, colS, wS, h, out1);

    // ---- layer 2 ----
    gcn_gemm_xWT<<<gemmGrid, 256, 0, stream>>>(out1, W1, h);
    gcn_aggregate<<<N, 256, 0, stream>>>(offsets, counts, colS, wS, h, out2);
}